// TemporalVoxelFusionV3_52312701665946
// MI455X (gfx1250) — compile-verified
//
#include <hip/hip_runtime.h>
#include <cstdint>
#include <cstddef>

#define EPSF 1e-6f

// Float atomic max via sign-split integer atomics (correct for mixed signs, no CAS loop).
// val >= 0: int-bit order == float order. val < 0: uint-bit order is reversed -> atomicMin.
__device__ __forceinline__ void atomicMaxF(float* addr, float val) {
  if (val >= 0.0f) {
    (void)atomicMax((int*)addr, __float_as_int(val));
  } else {
    (void)atomicMin((unsigned int*)addr, (unsigned int)__float_as_int(val));
  }
}

// Stage 256 feature rows (256*15 floats = 15360 B, 16B-aligned since 256*60 % 16 == 0)
// from global into LDS with gfx1250 async global->LDS b128 copies (ASYNCcnt path).
// Non-temporal hint: features are streamed once per pass and exceed L2 (252 MB > 192 MB),
// so keep them from evicting the hot M-sized accumulator lines.
__device__ __forceinline__ void stage_rows_async(const float* gbase, float* lds, int tid) {
  const unsigned lds0 = (unsigned)(uintptr_t)lds;          // low 32 bits == LDS offset
#pragma unroll
  for (int k = 0; k < 4; ++k) {
    unsigned c = (unsigned)tid + (unsigned)k * 256u;       // b128 chunk id
    if (c < 960u) {                                        // 15360 B / 16 B
      unsigned ofs = c * 16u;
      unsigned ldsaddr = lds0 + ofs;
      asm volatile("global_load_async_to_lds_b128 %0, %1, %2 th:TH_LOAD_NT"
                   :: "v"(ldsaddr), "v"(ofs), "s"(gbase) : "memory");
    }
  }
  asm volatile("s_wait_asynccnt 0x0" ::: "memory");
}

// Workspace layout (floats): [0,M) seg_max | [M,2M) seg_sum | [2M,3M) counts
//                            [3M,4M) max_opacity | [4M,18M) agg (M x 14, row-major)

__global__ void k_init(float* __restrict__ ws, int M) {
  int i = blockIdx.x * blockDim.x + threadIdx.x;
  if (i >= M * 18) return;
  // regions are contiguous: [0,M) seg_max and [3M,4M) maxop get -inf, rest 0 (no int div)
  bool neg_inf = (i < M) | ((i >= 3 * M) & (i < 4 * M));
  ws[i] = neg_inf ? -__builtin_huge_valf() : 0.0f;
}

// Pass 1 (N): sharp score per gaussian -> store into weights slot, atomic segment max.
// Only col 14 (g_time) is consumed, but at 60 B row stride every cache line is touched
// anyway; stage rows densely via async b128 instead of 15-line scattered b32 loads.
__global__ __launch_bounds__(256) void k_score(
    const float* __restrict__ feats, const int* __restrict__ vidx,
    const float* __restrict__ conf, const float* __restrict__ ttime,
    float* __restrict__ score_out, float* __restrict__ seg_max, int N) {
  __shared__ __align__(16) float lds[256 * 15];
  const int tid = threadIdx.x;
  const long long base = (long long)blockIdx.x * 256;
  const long long i = base + tid;
  const bool full = (base + 256) <= (long long)N;          // block-uniform

  if (full) stage_rows_async(feats + base * 15, lds, tid);
  __syncthreads();

  if (i < N) {
    float gt = full ? lds[tid * 15 + 14] : feats[(size_t)i * 15 + 14];
    float t0 = ttime[0];
    float g0 = fmaxf(conf[i], EPSF);
    float sigma = -2.3025850929940457f / (g0 * g0 + EPSF); // log(0.1)/(gamma0^2+eps)
    float dt = t0 - gt;
    float sc = logf(fmaxf(expf(sigma * dt * dt), EPSF)) / 0.07f;
    score_out[i] = sc;
    atomicMaxF(&seg_max[vidx[i]], sc);
  }
}

// Pass 2 (N): ex = exp(score - segmax); atomic seg_sum/counts/max_opacity and 14-wide
// weighted aggregate (un-normalized; divided by seg_sum in finalize).
__global__ __launch_bounds__(256) void k_accum(
    const float* __restrict__ feats, const int* __restrict__ vidx,
    float* __restrict__ score_ex, const float* __restrict__ seg_max,
    float* __restrict__ seg_sum, float* __restrict__ counts,
    float* __restrict__ maxop, float* __restrict__ agg, int N) {
  __shared__ __align__(16) float lds[256 * 15];
  const int tid = threadIdx.x;
  const long long base = (long long)blockIdx.x * 256;
  const long long i = base + tid;
  const bool full = (base + 256) <= (long long)N;          // block-uniform

  if (full) stage_rows_async(feats + base * 15, lds, tid);
  __syncthreads();

  if (i < N) {
    float f[15];
    if (full) {
#pragma unroll
      for (int j = 0; j < 15; ++j) f[j] = lds[tid * 15 + j];   // stride 15 -> conflict-free
    } else {
      const float* r = feats + (size_t)i * 15;
#pragma unroll
      for (int j = 0; j < 15; ++j) f[j] = r[j];
    }
    int v = vidx[i];
    float ex = expf(score_ex[i] - seg_max[v]);
    score_ex[i] = ex;                                      // overwrite score with ex
    atomicAdd(&seg_sum[v], ex);
    atomicAdd(&counts[v], 1.0f);
    atomicMaxF(&maxop[v], f[6]);                           // opacity
    float* arow = agg + (size_t)v * 14;
#pragma unroll
    for (int j = 0; j < 7; ++j)  atomicAdd(&arow[j], f[j] * ex);
#pragma unroll
    for (int j = 7; j < 10; ++j) atomicAdd(&arow[j], logf(fmaxf(f[j], EPSF)) * ex);
#pragma unroll
    for (int j = 10; j < 14; ++j) atomicAdd(&arow[j], f[j] * ex);
  }
}

// Pass 3 (N, light): weights = ex / seg_sum[idx]  (in place in the d_out weights slot)
__global__ void k_weights(const int* __restrict__ vidx, const float* __restrict__ seg_sum,
                          float* __restrict__ w, int N) {
  int i = blockIdx.x * blockDim.x + threadIdx.x;
  if (i >= N) return;
  w[i] = w[i] / seg_sum[vidx[i]];
}

// Pass 4 (M): finalize fused gaussians.
__global__ void k_final(const float* __restrict__ ws, float* __restrict__ out, int M) {
  int v = blockIdx.x * blockDim.x + threadIdx.x;
  if (v >= M) return;
  float s  = ws[(size_t)M + v];
  float c  = ws[(size_t)2 * M + v];
  float mo = ws[(size_t)3 * M + v];
  bool used = c > 0.0f;
  float inv = used ? 1.0f / s : 0.0f;                      // empty voxel: agg -> 0
  const float* arow = ws + (size_t)4 * M + (size_t)v * 14;
  float a[14];
#pragma unroll
  for (int j = 0; j < 14; ++j) a[j] = arow[j] * inv;
  float* o = out + (size_t)v * 15;
  o[0] = a[0]; o[1] = a[1]; o[2] = a[2];                   // pos
  o[3] = a[3]; o[4] = a[4]; o[5] = a[5];                   // color
  o[6] = 0.7f * (used ? mo : 0.0f) + 0.3f * a[6];          // opacity
  o[7] = expf(a[7]); o[8] = expf(a[8]); o[9] = expf(a[9]); // scales (empty -> exp(0)=1)
  float rn = sqrtf(a[10]*a[10] + a[11]*a[11] + a[12]*a[12] + a[13]*a[13]);
  float d = fmaxf(rn, 1e-6f);
  o[10] = a[10] / d; o[11] = a[11] / d; o[12] = a[12] / d; o[13] = a[13] / d;
  o[14] = 0.0f;                                            // dummy time
}

extern "C" void kernel_launch(void* const* d_in, const int* in_sizes, int n_in,
                              void* d_out, int out_size, void* d_ws, size_t ws_size,
                              hipStream_t stream) {
  (void)n_in; (void)ws_size;
  const float* feats = (const float*)d_in[0];   // [N,15] f32
  const int*   vidx  = (const int*)d_in[1];     // [N] i32
  const float* ttime = (const float*)d_in[3];   // [1] f32
  const float* conf  = (const float*)d_in[4];   // [N] f32
  const int N = in_sizes[1];
  const int M = (out_size - N) / 15;            // d_out = fused[M,15] ++ weights[N]

  float* out     = (float*)d_out;
  float* weights = out + (size_t)M * 15;        // also used as score/ex scratch
  float* ws      = (float*)d_ws;

  const int tb = 256;
  k_init   <<<(M * 18 + tb - 1) / tb, tb, 0, stream>>>(ws, M);
  k_score  <<<(N + tb - 1) / tb, tb, 0, stream>>>(feats, vidx, conf, ttime, weights, ws, N);
  k_accum  <<<(N + tb - 1) / tb, tb, 0, stream>>>(feats, vidx, weights, ws,
                                                  ws + (size_t)M, ws + (size_t)2 * M,
                                                  ws + (size_t)3 * M, ws + (size_t)4 * M, N);
  k_weights<<<(N + tb - 1) / tb, tb, 0, stream>>>(vidx, ws + (size_t)M, weights, N);
  k_final  <<<(M + tb - 1) / tb, tb, 0, stream>>>(ws, out, M);
}